// GNNModel_10007273799731
// MI455X (gfx1250) — compile-verified
//
#include <hip/hip_runtime.h>
#include <hip/hip_bf16.h>
#include <stdint.h>

#define N_MUS 100000
#define N_REC 50000
#define DD    128
#define NLAYER 3
#define NEDGE 1600000

typedef __attribute__((ext_vector_type(2))) float v2f;
typedef __attribute__((ext_vector_type(8))) float v8f;

// ---------------------------------------------------------------- zero fill
__global__ void zero_f32(float* __restrict__ p, long n4) {
  long i = (long)blockIdx.x * blockDim.x + threadIdx.x;
  if (i < n4) ((float4*)p)[i] = make_float4(0.f, 0.f, 0.f, 0.f);
}

// ---------------------------------------------------------------- edge degree
__global__ void count_edges(const int* __restrict__ dst, float* __restrict__ cnt, int E) {
  int e = blockIdx.x * blockDim.x + threadIdx.x;
  if (e < E) atomicAdd(&cnt[dst[e]], 1.0f);
}

// ------------------------------------------------- gather + atomic scatter-add
// one wave (32 lanes) per edge; 4 floats / lane = 128-wide feature row.
// agg buffers (<=51MB) live in the 192MB L2, so the f32 atomics resolve in L2.
__global__ void scatter_add(const float* __restrict__ xsrc, const int* __restrict__ src,
                            const int* __restrict__ dst, float* __restrict__ agg, int E) {
  int gw   = (int)(((long)blockIdx.x * blockDim.x + threadIdx.x) >> 5);
  int lane = threadIdx.x & 31;
  if (gw >= E) return;
  int s = src[gw];
  int d = dst[gw];
  const float4 v = *(const float4*)(xsrc + (long)s * DD + lane * 4);
  float* ap = agg + (long)d * DD + lane * 4;
  atomicAdd(ap + 0, v.x);
  atomicAdd(ap + 1, v.y);
  atomicAdd(ap + 2, v.z);
  atomicAdd(ap + 3, v.w);
}

// --------------------------------------------------------------- fused SAGE op
// out = l2norm((agg/max(cnt,1)) @ Wl + b + xdst @ Wr) [+ prev][relu]
// block: 256 threads = 8 waves; wave w -> 16-row strip, all 128 cols.
// WMMA f32 16x16x4, K accumulated in steps of 4 (full fp32 precision).
// Two passes over K (Wl then Wr) keep register pressure at 64 acc + 16 frag
// VGPRs so all 8 WMMAs per k-step issue back-to-back from preloaded LDS frags.
__launch_bounds__(256)
__global__ void sage_gemm(const float* __restrict__ agg, const float* __restrict__ cnt,
                          const float* __restrict__ xdst,
                          const float* __restrict__ Wl, const float* __restrict__ Wr,
                          const float* __restrict__ bias,
                          float* __restrict__ out, int Ndst, int doRelu, int doAccum) {
  extern __shared__ float smem[];      // 2 * 128*128 floats = 128 KB
  float* sWl = smem;
  float* sWr = smem + DD * DD;

  // Stage Wl,Wr into LDS with gfx1250 async global->LDS copies (no VGPR
  // round-trip; tracked by ASYNCcnt). 256 threads x 16B = 4KB per issue.
  {
    unsigned            ldsA = (unsigned)(uintptr_t)(void*)sWl + threadIdx.x * 16u;
    unsigned long long  gL   = (unsigned long long)(uintptr_t)Wl + threadIdx.x * 16u;
    unsigned long long  gR   = (unsigned long long)(uintptr_t)Wr + threadIdx.x * 16u;
    for (int i = 0; i < 16; ++i) {
      asm volatile("global_load_async_to_lds_b128 %0, %1, off"
                   :: "v"(ldsA + i * 4096u), "v"(gL + (unsigned)(i * 4096)) : "memory");
    }
    for (int i = 0; i < 16; ++i) {
      asm volatile("global_load_async_to_lds_b128 %0, %1, off"
                   :: "v"(ldsA + 65536u + i * 4096u), "v"(gR + (unsigned)(i * 4096)) : "memory");
    }
    asm volatile("s_wait_asynccnt 0" ::: "memory");
  }
  __syncthreads();

  const int wave    = threadIdx.x >> 5;
  const int lane    = threadIdx.x & 31;
  const int rowTile = blockIdx.x * 8 + wave;
  if (rowTile * 16 >= Ndst) return;          // wave-uniform exit, EXEC stays all-1
  const int  row0 = rowTile * 16;
  const int  m    = lane & 15;               // A row within tile; also B/D column
  const int  kh   = (lane >> 4) * 2;         // K sub-offset per half-wave
  const long grow = row0 + m;

  const float invc = 1.0f / fmaxf(cnt[grow], 1.0f);
  const float* aggRow = agg  + grow * DD;
  const float* xRow   = xdst + grow * DD;

  v8f acc[8];
  v8f z = {};
  for (int t = 0; t < 8; ++t) acc[t] = z;

  for (int pass = 0; pass < 2; ++pass) {
    const float* row = pass ? xRow : aggRow;
    const float* sW  = pass ? sWr  : sWl;
    const float  scl = pass ? 1.0f : invc;   // fold mean into the agg pass
    for (int k0 = 0; k0 < DD; k0 += 4) {
      v2f a;
      a[0] = row[k0 + kh] * scl;
      a[1] = row[k0 + kh + 1] * scl;
      const float* s0 = sW + (k0 + kh) * DD + m;
      const float* s1 = sW + (k0 + kh + 1) * DD + m;
      v2f bf[8];
#pragma unroll
      for (int t = 0; t < 8; ++t) {
        bf[t][0] = s0[t * 16];
        bf[t][1] = s1[t * 16];
      }
#pragma unroll
      for (int t = 0; t < 8; ++t)
        acc[t] = __builtin_amdgcn_wmma_f32_16x16x4_f32(false, a, false, bf[t],
                                                       (short)0, acc[t], false, false);
    }
  }

  // bias (per output column)
  for (int t = 0; t < 8; ++t) {
    float bv = bias[t * 16 + m];
    for (int v = 0; v < 8; ++v) acc[t][v] += bv;
  }

  // row-wise L2 norm: row r = v + 8*(lane>>4); its 128 elems live in the
  // 16-lane half-wave across the 8 column tiles.
  for (int v = 0; v < 8; ++v) {
    float ss = 0.f;
    for (int t = 0; t < 8; ++t) ss += acc[t][v] * acc[t][v];
    ss += __shfl_xor(ss, 1, 32);
    ss += __shfl_xor(ss, 2, 32);
    ss += __shfl_xor(ss, 4, 32);
    ss += __shfl_xor(ss, 8, 32);
    float scale = 1.0f / fmaxf(sqrtf(ss), 1e-12f);
    for (int t = 0; t < 8; ++t) acc[t][v] *= scale;
  }

  // epilogue: optional hetero accumulate + relu, store
  const int hi = lane >> 4;
  for (int v = 0; v < 8; ++v) {
    long r = row0 + v + 8 * hi;
    float* orow = out + r * DD;
    for (int t = 0; t < 8; ++t) {
      int col = t * 16 + m;
      float val = acc[t][v];
      if (doAccum) val += orow[col];
      if (doRelu)  val = fmaxf(val, 0.0f);
      orow[col] = val;
    }
  }
}

extern "C" void kernel_launch(void* const* d_in, const int* in_sizes, int n_in,
                              void* d_out, int out_size, void* d_ws, size_t ws_size,
                              hipStream_t stream) {
  (void)in_sizes; (void)n_in; (void)out_size; (void)ws_size;
  const float* x_mus = (const float*)d_in[0];
  const float* x_rec = (const float*)d_in[1];
  const float* Wl    = (const float*)d_in[2];   // [L,3,D,D]
  const float* Wr    = (const float*)d_in[3];   // [L,3,D,D]
  const float* b     = (const float*)d_in[4];   // [L,3,D]
  const int*   e_mr  = (const int*)d_in[5];     // [2,E]
  const int*   e_rm  = (const int*)d_in[6];
  const int*   e_mm  = (const int*)d_in[7];

  float* ws     = (float*)d_ws;
  float* agg    = ws;                              // N_MUS*DD (reused per relation)
  float* cnt_mr = agg + (long)N_MUS * DD;          // N_REC
  float* cnt_rm = cnt_mr + N_REC;                  // N_MUS
  float* cnt_mm = cnt_rm + N_MUS;                  // N_MUS
  float* xm_a   = cnt_mm + N_MUS;                  // N_MUS*DD
  float* xm_b   = xm_a + (long)N_MUS * DD;         // N_MUS*DD
  float* xr_a   = xm_b + (long)N_MUS * DD;         // N_REC*DD
  float* xr_b   = xr_a + (long)N_REC * DD;         // N_REC*DD

  float* out_mus = (float*)d_out;
  float* out_rec = out_mus + (long)N_MUS * DD;

  const int    E      = NEDGE;
  const size_t SHMEM  = 2 * DD * DD * sizeof(float);
  const int    cblks  = (E + 255) / 256;
  const int    sblks  = (E + 7) / 8;               // 8 edges (waves) per block
  const int    gmus   = (N_MUS / 16 + 7) / 8;
  const int    grec   = (N_REC / 16 + 7) / 8;

  auto zero = [&](float* p, long nflt) {
    long n4 = nflt / 4;
    zero_f32<<<(int)((n4 + 255) / 256), 256, 0, stream>>>(p, n4);
  };

  // in-degrees are layer-invariant: compute once per call
  zero(cnt_mr, N_REC); zero(cnt_rm, N_MUS); zero(cnt_mm, N_MUS);
  count_edges<<<cblks, 256, 0, stream>>>(e_mr + E, cnt_mr, E);
  count_edges<<<cblks, 256, 0, stream>>>(e_rm + E, cnt_rm, E);
  count_edges<<<cblks, 256, 0, stream>>>(e_mm + E, cnt_mm, E);

  const float* xm = x_mus;
  const float* xr = x_rec;
  for (int l = 0; l < NLAYER; ++l) {
    float* xm_o = (l == 0) ? xm_a : (l == 1) ? xm_b : out_mus;
    float* xr_o = (l == 0) ? xr_a : (l == 1) ? xr_b : out_rec;
    int relu = (l < NLAYER - 1) ? 1 : 0;
    const float* Wl_l = Wl + (long)l * 3 * DD * DD;
    const float* Wr_l = Wr + (long)l * 3 * DD * DD;
    const float* b_l  = b  + (long)l * 3 * DD;

    // rec <- (mus -> rec)
    zero(agg, (long)N_REC * DD);
    scatter_add<<<sblks, 256, 0, stream>>>(xm, e_mr, e_mr + E, agg, E);
    sage_gemm<<<grec, 256, SHMEM, stream>>>(agg, cnt_mr, xr, Wl_l, Wr_l, b_l,
                                            xr_o, N_REC, relu, 0);

    // mus <- (rec -> mus): store normalized result, no relu yet
    zero(agg, (long)N_MUS * DD);
    scatter_add<<<sblks, 256, 0, stream>>>(xr, e_rm, e_rm + E, agg, E);
    sage_gemm<<<gmus, 256, SHMEM, stream>>>(agg, cnt_rm, xm,
                                            Wl_l + DD * DD, Wr_l + DD * DD, b_l + DD,
                                            xm_o, N_MUS, 0, 0);

    // mus += (mus -> mus): accumulate, then relu on the hetero sum
    zero(agg, (long)N_MUS * DD);
    scatter_add<<<sblks, 256, 0, stream>>>(xm, e_mm, e_mm + E, agg, E);
    sage_gemm<<<gmus, 256, SHMEM, stream>>>(agg, cnt_mm, xm,
                                            Wl_l + 2 * DD * DD, Wr_l + 2 * DD * DD,
                                            b_l + 2 * DD,
                                            xm_o, N_MUS, relu, 1);

    xm = xm_o; xr = xr_o;
  }
}